// FastTextItout_42554535969503
// MI455X (gfx1250) — compile-verified
//
#include <hip/hip_runtime.h>

typedef float v2f __attribute__((ext_vector_type(2)));
typedef float v8f __attribute__((ext_vector_type(8)));

#define NB 16384
#define TT 64
#define EMBD 16
#define HID 1024

__global__ __launch_bounds__(256, 2)
void fasttext_lif_wmma(const int* __restrict__ xis,
                       const float* __restrict__ emb,
                       const float* __restrict__ lif_w,
                       const float* __restrict__ lif_b,
                       const float* __restrict__ fc_w,
                       const float* __restrict__ fc_b,
                       const float* __restrict__ alpha_p,
                       const float* __restrict__ beta_p,
                       float* __restrict__ out)
{
    // lif_w staged pair-interleaved: s_lw2[(k/2)*HID + n] = {lif_w[k][n], lif_w[k+1][n]} (k even)
    __shared__ float2 s_lw2[(EMBD / 2) * HID];
    __shared__ float2 s_comb[HID];   // {lif_b[n], fc_w[n]}

    const int tid   = threadIdx.x;
    const int lane  = tid & 31;
    const int wave  = tid >> 5;
    const int m     = lane & 15;     // row-in-tile / column-in-tile index
    const int thalf = lane >> 4;     // which half of the wave

    const float alpha = alpha_p[0];
    const float beta  = beta_p[0];

    // ---- Stage weights into LDS ----
    for (int p = tid; p < (EMBD / 2) * HID; p += 256) {
        int k2 = p >> 10;            // 0..7
        int n  = p & (HID - 1);
        float x = lif_w[(2 * k2) * HID + n];
        float y = lif_w[(2 * k2 + 1) * HID + n];
        s_lw2[k2 * HID + n] = make_float2(x, y);
    }
    for (int n = tid; n < HID; n += 256)
        s_comb[n] = make_float2(lif_b[n], fc_w[n]);
    __syncthreads();

    // ---- Phase 1: weighted embedding bag (collapsed LIF scan) ----
    const int rowbase = blockIdx.x * 128 + wave * 16;
    const int brow    = rowbase + m;
    const int* xrow   = xis + (size_t)brow * TT + thalf * 32;

    const float a2  = alpha * alpha;
    const float a4  = a2 * a2;
    const float a8  = a4 * a4;
    const float a16 = a8 * a8;
    const float a32 = a16 * a16;

    // lanes 0-15 handle t=0..31 (weights alpha^63..alpha^32),
    // lanes 16-31 handle t=32..63 (weights alpha^31..alpha^0)
    float w = thalf ? 1.0f : a32;

    float acc[16];
#pragma unroll
    for (int d = 0; d < 16; ++d) acc[d] = 0.0f;

#pragma unroll 4
    for (int t = 31; t >= 0; --t) {
        int id = xrow[t];
        const float4* e = (const float4*)(emb + (size_t)id * EMBD);
        float4 e0 = e[0], e1 = e[1], e2 = e[2], e3 = e[3];
        acc[0]  = fmaf(w, e0.x, acc[0]);
        acc[1]  = fmaf(w, e0.y, acc[1]);
        acc[2]  = fmaf(w, e0.z, acc[2]);
        acc[3]  = fmaf(w, e0.w, acc[3]);
        acc[4]  = fmaf(w, e1.x, acc[4]);
        acc[5]  = fmaf(w, e1.y, acc[5]);
        acc[6]  = fmaf(w, e1.z, acc[6]);
        acc[7]  = fmaf(w, e1.w, acc[7]);
        acc[8]  = fmaf(w, e2.x, acc[8]);
        acc[9]  = fmaf(w, e2.y, acc[9]);
        acc[10] = fmaf(w, e2.z, acc[10]);
        acc[11] = fmaf(w, e2.w, acc[11]);
        acc[12] = fmaf(w, e3.x, acc[12]);
        acc[13] = fmaf(w, e3.y, acc[13]);
        acc[14] = fmaf(w, e3.z, acc[14]);
        acc[15] = fmaf(w, e3.w, acc[15]);
        w *= alpha;
    }

    // combine the two timestep halves: every lane gets the full 16-dim row
#pragma unroll
    for (int d = 0; d < 16; ++d) acc[d] += __shfl_xor(acc[d], 16);

    // A operand for V_WMMA_F32_16X16X4_F32: lanes 0-15 hold K={4g,4g+1}, lanes 16-31 K={4g+2,4g+3}
    v2f aop[4];
#pragma unroll
    for (int g = 0; g < 4; ++g) {
        aop[g].x = thalf ? acc[4 * g + 2] : acc[4 * g + 0];
        aop[g].y = thalf ? acc[4 * g + 3] : acc[4 * g + 1];
    }

    // geometric sum s = (1-a^64)/(1-a) via exact telescoping product
    const float sgeo = (1.0f + alpha) * (1.0f + a2) * (1.0f + a4) *
                       (1.0f + a8) * (1.0f + a16) * (1.0f + a32);

    // ---- Phase 2: [16x16] @ [16x1024] fused with bias + relu + fc dot ----
    float racc[8];
#pragma unroll
    for (int r = 0; r < 8; ++r) racc[r] = 0.0f;

    for (int n0 = 0; n0 < HID; n0 += 16) {
        const int n = n0 + m;   // this lane's column
        // B operand: lanes 0-15 hold K={4g,4g+1}, lanes 16-31 K={4g+2,4g+3}, N=lane&15
        v2f bop[4];
#pragma unroll
        for (int g = 0; g < 4; ++g) {
            float2 t = s_lw2[(2 * g + thalf) * HID + n];
            bop[g].x = t.x;
            bop[g].y = t.y;
        }

        v8f c = {};
        c = __builtin_amdgcn_wmma_f32_16x16x4_f32(false, aop[0], false, bop[0], (short)0, c, false, false);
        c = __builtin_amdgcn_wmma_f32_16x16x4_f32(false, aop[1], false, bop[1], (short)0, c, false, false);
        c = __builtin_amdgcn_wmma_f32_16x16x4_f32(false, aop[2], false, bop[2], (short)0, c, false, false);
        c = __builtin_amdgcn_wmma_f32_16x16x4_f32(false, aop[3], false, bop[3], (short)0, c, false, false);

        float2 cb   = s_comb[n];
        float  bias = sgeo * (cb.x + beta);
#pragma unroll
        for (int r = 0; r < 8; ++r) {
            float v = c[r] + bias;
            v = v > 0.0f ? v : 0.0f;          // relu
            racc[r] = fmaf(v, cb.y, racc[r]); // * fc_w[n], accumulate over columns
        }
    }

    // ---- Row reduction across the 16 lanes of each half (C layout: lane=col, VGPR r=row) ----
#pragma unroll
    for (int r = 0; r < 8; ++r) {
        racc[r] += __shfl_xor(racc[r], 8);
        racc[r] += __shfl_xor(racc[r], 4);
        racc[r] += __shfl_xor(racc[r], 2);
        racc[r] += __shfl_xor(racc[r], 1);
    }

    if (m == 0) {
        const float fcb = fc_b[0];
        const int ob = rowbase + thalf * 8;   // lanes 0-15 hold rows 0-7, lanes 16-31 rows 8-15
#pragma unroll
        for (int r = 0; r < 8; ++r)
            out[ob + r] = racc[r] + fcb;
    }
}

extern "C" void kernel_launch(void* const* d_in, const int* in_sizes, int n_in,
                              void* d_out, int out_size, void* d_ws, size_t ws_size,
                              hipStream_t stream) {
    const int*   xis   = (const int*)  d_in[0];
    const float* emb   = (const float*)d_in[1];
    const float* lif_w = (const float*)d_in[2];
    const float* lif_b = (const float*)d_in[3];
    const float* fc_w  = (const float*)d_in[4];
    const float* fc_b  = (const float*)d_in[5];
    const float* alpha = (const float*)d_in[6];
    const float* beta  = (const float*)d_in[7];
    float*       out   = (float*)d_out;

    dim3 grid(NB / 128);   // 128 blocks, each: 8 waves x 16 rows
    dim3 block(256);
    fasttext_lif_wmma<<<grid, block, 0, stream>>>(xis, emb, lif_w, lif_b, fc_w,
                                                  fc_b, alpha, beta, out);
}